// VariationalGCNEncoder_43722767073852
// MI455X (gfx1250) — compile-verified
//
#include <hip/hip_runtime.h>
#include <hip/hip_bf16.h>

typedef __attribute__((ext_vector_type(2))) float v2f;
typedef __attribute__((ext_vector_type(8))) float v8f;

// ---------------------------------------------------------------------------
// Utility kernels
// ---------------------------------------------------------------------------
__global__ void zero_kernel(float* __restrict__ p, long long n) {
  long long i = (long long)blockIdx.x * blockDim.x + threadIdx.x;
  if (i < n) p[i] = 0.0f;
}

__global__ void degree_kernel(const long long* __restrict__ ei, float* __restrict__ deg, int E) {
  int i = blockIdx.x * blockDim.x + threadIdx.x;
  if (i < E) {
    int d = (int)ei[E + i];  // dst row of edge_index [2,E]
    atomicAdd(&deg[d], 1.0f);
  }
}

__global__ void rsqrt_kernel(float* __restrict__ dinv, int n) {
  int i = blockIdx.x * blockDim.x + threadIdx.x;
  if (i < n) dinv[i] = rsqrtf(dinv[i] + 1.0f);  // +1 for self loop
}

// ---------------------------------------------------------------------------
// Pack B[K,N] (row-major) into WMMA fragment order: Bp[r][c] = {B[2r][c], B[2r+1][c]}
// so a lane's B fragment (two consecutive K values, one column) is one b64 load.
// ---------------------------------------------------------------------------
__global__ void pack_B_kernel(const float* __restrict__ B, float2* __restrict__ Bp,
                              int K, int N) {
  int i = blockIdx.x * blockDim.x + threadIdx.x;
  int tot = (K >> 1) * N;
  if (i < tot) {
    int r = i / N, c = i % N;
    float2 v;
    v.x = B[(size_t)(2 * r) * N + c];
    v.y = B[(size_t)(2 * r + 1) * N + c];
    Bp[i] = v;
  }
}

// ---------------------------------------------------------------------------
// FP32 WMMA GEMM: C[M,N] = A[M,K] * B[K,N], row-major; packed B staged in LDS.
// One wave computes a 16(M) x (16*NSUB)(N) tile; one A fragment feeds NSUB WMMAs.
// V_WMMA_F32_16X16X4_F32 fragment layouts (wave32):
//   A 16x4 : lanes 0-15 row M=l  hold K=k+0,k+1 ; lanes 16-31 row M=l-16 hold K=k+2,k+3
//   B 4x16 : lanes 0-15 col N=l  hold K=k+0,k+1 ; lanes 16-31 col N=l-16 hold K=k+2,k+3
//   C/D    : VGPR i -> row i + 8*(lane>=16), col = lane&15
// ---------------------------------------------------------------------------
template <int NSUB>
__global__ void gemm_wmma_f32_lds(const float* __restrict__ A, const float2* __restrict__ Bp,
                                  float* __restrict__ C, int M, int K, int N) {
  extern __shared__ float2 sB[];  // (K/2) * N packed pairs  (<= 64 KB)
  const int tid = threadIdx.x;
  const int kn2 = (K >> 1) * N;
  for (int i = tid; i < kn2; i += blockDim.x) sB[i] = Bp[i];
  __syncthreads();

  const int wave = blockIdx.x * (blockDim.x >> 5) + (tid >> 5);
  const int lane = tid & 31;
  const int ngrp = N / (16 * NSUB);  // column groups (1 in our configs)
  const int mt   = wave / ngrp;
  const int cg   = wave % ngrp;
  const int row0 = mt * 16;
  if (row0 >= M) return;
  const int col0 = cg * 16 * NSUB;

  const int half = lane >> 4;  // 0: lanes 0-15, 1: lanes 16-31
  const int l16  = lane & 15;

  v8f acc[NSUB];
#pragma unroll
  for (int t = 0; t < NSUB; ++t) acc[t] = (v8f){};

  const float* __restrict__ Arow = A + (size_t)(row0 + l16) * K + half * 2;

  for (int k = 0; k < K; k += 4) {
    v2f a;
    a.x = Arow[k];
    a.y = Arow[k + 1];
    const float2* bp = sB + ((k >> 1) + half) * N + col0 + l16;
#pragma unroll
    for (int t = 0; t < NSUB; ++t) {
      float2 bv = bp[t * 16];
      v2f b;
      b.x = bv.x;
      b.y = bv.y;
      acc[t] = __builtin_amdgcn_wmma_f32_16x16x4_f32(false, a, false, b, (short)0,
                                                     acc[t], false, false);
    }
  }

#pragma unroll
  for (int i = 0; i < 8; ++i) {
    int r = row0 + i + half * 8;
    if (r < M) {
      float* crow = C + (size_t)r * N + col0 + l16;
#pragma unroll
      for (int t = 0; t < NSUB; ++t) crow[t * 16] = acc[t][i];
    }
  }
}

// ---------------------------------------------------------------------------
// Edge scatter: one wave per edge, lanes stride over channels.
//   agg[dst, c] += h[src, c] * dinv[src] * dinv[dst]
// Prefetch next edge's source row to hide gather latency.
// ---------------------------------------------------------------------------
__global__ void scatter_edges(const long long* __restrict__ ei,
                              const float* __restrict__ h,
                              const float* __restrict__ dinv,
                              float* __restrict__ agg, int E, int C) {
  const int wid    = (blockIdx.x * blockDim.x + threadIdx.x) >> 5;
  const int lane   = threadIdx.x & 31;
  const int nwaves = (gridDim.x * blockDim.x) >> 5;

  for (int e = wid; e < E; e += nwaves) {
    const int e2 = e + nwaves;
    if (e2 < E) {
      const int s2 = (int)ei[e2];
      __builtin_prefetch(h + (size_t)s2 * C + lane, 0, 1);  // global_prefetch_b8
    }
    const int s = (int)ei[e];
    const int d = (int)ei[E + e];
    const float norm = dinv[s] * dinv[d];
    const float* __restrict__ hs = h + (size_t)s * C;
    float* __restrict__ ad = agg + (size_t)d * C;
#pragma unroll 4
    for (int c = lane; c < C; c += 32) {
      atomicAdd(&ad[c], hs[c] * norm);
    }
  }
}

// ---------------------------------------------------------------------------
// Finalize: out = agg + h * dinv^2 + bias  (optional ReLU). In-place safe.
// ---------------------------------------------------------------------------
__global__ void finalize_kernel(const float* __restrict__ agg, const float* __restrict__ h,
                                const float* __restrict__ dinv, const float* __restrict__ bias,
                                float* __restrict__ out, int Nn, int C, int do_relu) {
  int i = blockIdx.x * blockDim.x + threadIdx.x;
  if (i >= Nn * C) return;
  int node = i / C;
  int c    = i % C;
  float di = dinv[node];
  float v  = agg[i] + h[i] * di * di + bias[c];
  out[i] = do_relu ? fmaxf(v, 0.0f) : v;
}

// ---------------------------------------------------------------------------
// One GCN layer: gemmOut = X@W ; agg = scatter ; out = agg + self + bias (+relu)
// ---------------------------------------------------------------------------
static void run_gcn_layer(const float* X, const float* W, const float* bias,
                          float* gemmOut, float* aggOut, const float* dinv, float2* BpBuf,
                          const long long* ei, int Nn, int E, int K, int C,
                          bool relu, hipStream_t stream) {
  // Pack weights into fragment order
  const int tot = (K >> 1) * C;
  pack_B_kernel<<<(tot + 255) / 256, 256, 0, stream>>>(W, BpBuf, K, C);

  // GEMM (8 waves / block; each wave does 16 rows x full C columns)
  const size_t ldsBytes = (size_t)tot * sizeof(float2);
  const int waves  = (Nn + 15) / 16;
  const int blocks = (waves + 7) / 8;
  if (C == 128)
    gemm_wmma_f32_lds<8><<<blocks, 256, ldsBytes, stream>>>(X, BpBuf, gemmOut, Nn, K, C);
  else
    gemm_wmma_f32_lds<4><<<blocks, 256, ldsBytes, stream>>>(X, BpBuf, gemmOut, Nn, K, C);

  const long long total = (long long)Nn * C;
  zero_kernel<<<(int)((total + 255) / 256), 256, 0, stream>>>(aggOut, total);

  scatter_edges<<<2048, 256, 0, stream>>>(ei, gemmOut, dinv, aggOut, E, C);

  finalize_kernel<<<(int)((total + 255) / 256), 256, 0, stream>>>(
      aggOut, gemmOut, dinv, bias, aggOut, Nn, C, relu ? 1 : 0);
}

// ---------------------------------------------------------------------------
// kernel_launch
// d_in order: x, edge_index(int64), W1n, b1n, W2n, b2n, W1e, b1e, W2e, b2e
// d_out: [mu (N*64) | logstd (N*64)] fp32
// ws: dinv[N] | bufA[N*128] | bufB[N*128] | packedB[<=16384 floats]
// ---------------------------------------------------------------------------
extern "C" void kernel_launch(void* const* d_in, const int* in_sizes, int n_in,
                              void* d_out, int out_size, void* d_ws, size_t ws_size,
                              hipStream_t stream) {
  const float*     x   = (const float*)d_in[0];
  const long long* ei  = (const long long*)d_in[1];
  const float*     W1n = (const float*)d_in[2];
  const float*     b1n = (const float*)d_in[3];
  const float*     W2n = (const float*)d_in[4];
  const float*     b2n = (const float*)d_in[5];
  const float*     W1e = (const float*)d_in[6];
  const float*     b1e = (const float*)d_in[7];
  const float*     W2e = (const float*)d_in[8];
  const float*     b2e = (const float*)d_in[9];

  const int IN_C = 128, HID = 128, OUT_C = 64;
  const int Nn = in_sizes[0] / IN_C;   // 100000
  const int E  = in_sizes[1] / 2;      // 3200000

  float* ws   = (float*)d_ws;
  float* dinv = ws;
  float* bufA = ws + Nn;                      // GEMM outputs (N x 128 max)
  float* bufB = bufA + (size_t)Nn * HID;      // agg / activations (N x 128 max)
  size_t bpOff = (((size_t)Nn + 2 * (size_t)Nn * HID) + 3) & ~(size_t)3;  // 16B align
  float2* BpBuf = (float2*)(ws + bpOff);      // packed weights (<=64 KB)

  float* mu     = (float*)d_out;
  float* logstd = mu + (size_t)Nn * OUT_C;

  // Degrees -> dinv (computed once, shared by all 4 layers)
  zero_kernel<<<(Nn + 255) / 256, 256, 0, stream>>>(dinv, Nn);
  degree_kernel<<<(E + 255) / 256, 256, 0, stream>>>(ei, dinv, E);
  rsqrt_kernel<<<(Nn + 255) / 256, 256, 0, stream>>>(dinv, Nn);

  // Layer 1: h = relu(gcn(x, W1n, b1n))           -> bufB [N,128]
  run_gcn_layer(x, W1n, b1n, bufA, bufB, dinv, BpBuf, ei, Nn, E, IN_C, HID, true, stream);
  // Layer 2: mu = gcn(h, W2n, b2n)                -> d_out[0:N*64]
  run_gcn_layer(bufB, W2n, b2n, bufA, mu, dinv, BpBuf, ei, Nn, E, HID, OUT_C, false, stream);
  // Layer 3: e = relu(gcn(mu, W1e, b1e))          -> bufB [N,128]
  run_gcn_layer(mu, W1e, b1e, bufA, bufB, dinv, BpBuf, ei, Nn, E, OUT_C, HID, true, stream);
  // Layer 4: logstd = gcn(e, W2e, b2e)            -> d_out[N*64:2*N*64]
  run_gcn_layer(bufB, W2e, b2e, bufA, logstd, dinv, BpBuf, ei, Nn, E, HID, OUT_C, false, stream);
}